// DeformConv_1Dto2D_12833362280762
// MI455X (gfx1250) — compile-verified
//
#include <hip/hip_runtime.h>
#include <hip/hip_bf16.h>

typedef float v2f __attribute__((ext_vector_type(2)));
typedef float v8f __attribute__((ext_vector_type(8)));
typedef unsigned int v4u __attribute__((ext_vector_type(4)));
typedef int v4i __attribute__((ext_vector_type(4)));
typedef int v8i __attribute__((ext_vector_type(8)));

#define LSIG   4096
#define NCH    16
#define OUTC   64
#define KS     7
#define ROWF   17                     // LDS row pitch in floats: 16 channels + 1 pad DWORD
#define CHUNKS 16
#define LCHUNK (LSIG / CHUNKS)        // 256
#define THREADS 512
#define NWAVES (THREADS / 32)         // 16
#define PERWAVE (LCHUNK / NWAVES)     // 16
#define LDS_BYTES ((LSIG + 2) * ROWF * 4)   // rows q=-1..4096 -> 278664 B < 320 KB

__global__ __launch_bounds__(THREADS, 1)
void deform_conv1d_wmma(const float* __restrict__ x,
                        const float* __restrict__ p_w, const float* __restrict__ p_b,
                        const float* __restrict__ m_w, const float* __restrict__ m_b,
                        const float* __restrict__ c_w, const float* __restrict__ c_b,
                        float* __restrict__ out)
{
    extern __shared__ float lds[];   // [LSIG+2 rows][ROWF]; row(q) = (q+1)*ROWF, +c

    const int blk   = blockIdx.x;
    const int b     = blk / CHUNKS;
    const int chunk = blk % CHUNKS;
    const int t     = threadIdx.x;

    // ---- fill LDS with x[b, :, :] as [l][c] rows, pitch 17 DWORDs ----------
#if __has_builtin(__builtin_amdgcn_tensor_load_to_lds)
    if (t < 32) {   // wave 0 issues one TDM op for the whole 256 KB tile
        const unsigned long long gaddr =
            (unsigned long long)(uintptr_t)x + (unsigned long long)b * (LSIG * NCH * 4);
        const unsigned int ldsaddr = (unsigned int)(uintptr_t)lds + ROWF * 4; // row q=0

        v4u g0;
        g0.x = 1u;                                         // count=1, user D#
        g0.y = ldsaddr;                                    // lds_addr (bytes)
        g0.z = (unsigned int)(gaddr & 0xFFFFFFFFu);        // global_addr[31:0]
        g0.w = (unsigned int)((gaddr >> 32) & 0x01FFFFFFu) // global_addr[56:32]
             | (2u << 30);                                 // type = 2 (image)

        v8i g1;
        g1[0] = (2 << 16)      // data_size = 4 bytes
              | (1 << 20)      // pad_enable
              | (3 << 22);     // pad_interval: 16 DWORDs ; pad_amount(31:25)=0 -> 1 DWORD
        g1[1] = (NCH  << 16);  // tensor_dim0 = 16   (bits 79:48, low half)
        g1[2] = (LSIG << 16);  // tensor_dim1 = 4096 (bits 111:80, low half)
        g1[3] = (NCH  << 16);  // tensor_dim1 hi = 0 | tile_dim0 = 16 (bits 127:112)
        g1[4] = LSIG;          // tile_dim1 = 4096 | tile_dim2 = 0
        g1[5] = NCH;           // tensor_dim0_stride = 16 (low 32 of bits 207:160)
        g1[6] = 0;
        g1[7] = 0;

        v4i gz4 = {0, 0, 0, 0};                            // 2D tensor: groups 2/3 unused
        v8i gz8 = {0, 0, 0, 0, 0, 0, 0, 0};
        __builtin_amdgcn_tensor_load_to_lds(g0, g1, gz4, gz4, gz8, 0);
        __builtin_amdgcn_s_wait_tensorcnt(0);
    }
    if (t < ROWF)                    lds[t] = 0.0f;                            // row q=-1
    else if (t >= 32 && t < 32+ROWF) lds[(LSIG + 1) * ROWF + (t - 32)] = 0.0f; // row q=4096
#else
    {
        const float* xb = x + (size_t)b * (LSIG * NCH);
        const int cg = (t & 3) * 4;
        const int p0 = t >> 2;
        #pragma unroll 4
        for (int pass = 0; pass < LSIG / (THREADS / 4); ++pass) {
            const int pos = p0 + pass * (THREADS / 4);
            const float4 v = *reinterpret_cast<const float4*>(xb + (size_t)pos * NCH + cg);
            float* r = lds + (size_t)(pos + 1) * ROWF + cg;
            r[0] = v.x; r[1] = v.y; r[2] = v.z; r[3] = v.w;
        }
        if (t < ROWF)                    lds[t] = 0.0f;
        else if (t >= 32 && t < 32+ROWF) lds[(LSIG + 1) * ROWF + (t - 32)] = 0.0f;
    }
#endif
    __syncthreads();

    const int lane = t & 31;
    const int wave = t >> 5;
    const int h    = lane >> 4;      // lane half: selects k-pair
    const int nidx = lane & 15;      // N index (channel c) / A row (oc within tile)

    // per-lane k set: step0 -> {2h, 2h+1}, step1 -> {2h+4, 2h+5 (==7 invalid)}
    int kid[4] = { 2*h, 2*h + 1, 2*h + 4, 2*h + 5 };

    // ---- deform-conv weights for this lane's 4 k values --------------------
    float pw[4][3], pbv[4], mw[4][3], mbv[4];
    bool  kval[4];
    #pragma unroll
    for (int i = 0; i < 4; ++i) {
        const bool valid = kid[i] < KS;
        kval[i] = valid;
        const int kk = valid ? kid[i] : 0;
        #pragma unroll
        for (int j = 0; j < 3; ++j) {
            pw[i][j] = valid ? p_w[kk * 3 + j] : 0.0f;
            mw[i][j] = valid ? m_w[kk * 3 + j] : 0.0f;
        }
        pbv[i] = valid ? p_b[kk] : 0.0f;
        mbv[i] = valid ? m_b[kk] : 0.0f;
    }

    // ---- A fragments: c_w tiles (16 oc x 4 k), constant per wave -----------
    // A layout (f32 16x4): lanes0-15 VGPR0=K0,VGPR1=K1 ; lanes16-31 K2,K3
    v2f a0[4], a1[4];
    #pragma unroll
    for (int tt = 0; tt < 4; ++tt) {
        const int oc = tt * 16 + nidx;
        a0[tt].x = c_w[oc * KS + 2*h];
        a0[tt].y = c_w[oc * KS + 2*h + 1];
        a1[tt].x = c_w[oc * KS + 2*h + 4];
        a1[tt].y = (2*h + 5 < KS) ? c_w[oc * KS + 2*h + 5] : 0.0f;
    }

    // ---- bias-initialized accumulator fragments ----------------------------
    // C/D layout: element (lane,v): M = v + 8h (= oc in tile), N = lane&15 (= c)
    v8f cbf[4];
    #pragma unroll
    for (int tt = 0; tt < 4; ++tt)
        #pragma unroll
        for (int v = 0; v < 8; ++v)
            cbf[tt][v] = c_b[tt * 16 + v + 8*h];

    const int    c   = nidx;             // this lane's channel for B building
    const float* col = lds + c;          // column c; value(q) = col[(q+1)*ROWF]
    // out[b,oc,l,c] flat = b*4194304 + oc*65536 + l*16 + c
    float* const outb = out + (size_t)b * (OUTC * LSIG * NCH) + c;

    const int lbase = chunk * LCHUNK + wave * PERWAVE;

    #pragma unroll 2
    for (int li = 0; li < PERWAVE; ++li) {
        const int l  = lbase + li;
        const int rb = l * ROWF;                 // row of q = l-1
        const float xm1 = col[rb];               // zero pad row handles l==0
        const float x0  = col[rb + ROWF];
        const float xp1 = col[rb + 2 * ROWF];    // zero pad row handles l==4095

        float xo[4];
        #pragma unroll
        for (int i = 0; i < 4; ++i) {
            const float off = pw[i][0]*xm1 + pw[i][1]*x0 + pw[i][2]*xp1 + pbv[i];
            const float mm  = mw[i][0]*xm1 + mw[i][1]*x0 + mw[i][2]*xp1 + mbv[i];
            // sigmoid via v_exp_f32 + v_rcp_f32 (avoid IEEE div expansion)
            const float sg  = __builtin_amdgcn_rcpf(1.0f + __expf(-mm));
            const float p   = (float)(l + 1) + (float)(kid[i] - 3) + off;
            float qf = floorf(p);
            qf = fminf(fmaxf(qf, 0.0f), (float)(LSIG - 1));
            const int qlt = (int)qf;
            int qrb = qlt + 1; if (qrb > LSIG - 1) qrb = LSIG - 1;
            const float pc  = fminf(fmaxf(p, 0.0f), (float)(LSIG - 1));
            const float glt = 1.0f + (qf - pc);
            const float grb = 1.0f - ((float)qrb - pc);
            const float val = glt * col[(qlt + 1) * ROWF] + grb * col[(qrb + 1) * ROWF];
            xo[i] = kval[i] ? (val * sg) : 0.0f;
        }

        // B fragments (f32 4x16): lanes0-15 VGPR0=K0,VGPR1=K1 ; lanes16-31 K2,K3
        v2f B0; B0.x = xo[0]; B0.y = xo[1];
        v2f B1; B1.x = xo[2]; B1.y = xo[3];

        #pragma unroll
        for (int tt = 0; tt < 4; ++tt) {
            v8f acc = cbf[tt];
            acc = __builtin_amdgcn_wmma_f32_16x16x4_f32(false, a0[tt], false, B0,
                                                        (short)0, acc, false, false);
            acc = __builtin_amdgcn_wmma_f32_16x16x4_f32(false, a1[tt], false, B1,
                                                        (short)0, acc, false, false);
            // store: oc = tt*16 + v + 8h, addr = oc*65536 + l*16 (+c in outb)
            float* o = outb + (size_t)l * NCH + (size_t)(tt * 16 + 8*h) * (LSIG * NCH);
            #pragma unroll
            for (int v = 0; v < 8; ++v)
                o[(size_t)v * (LSIG * NCH)] = acc[v];
        }
    }
}

extern "C" void kernel_launch(void* const* d_in, const int* in_sizes, int n_in,
                              void* d_out, int out_size, void* d_ws, size_t ws_size,
                              hipStream_t stream) {
    const float* x   = (const float*)d_in[0];
    const float* p_w = (const float*)d_in[1];
    const float* p_b = (const float*)d_in[2];
    const float* m_w = (const float*)d_in[3];
    const float* m_b = (const float*)d_in[4];
    const float* c_w = (const float*)d_in[5];
    const float* c_b = (const float*)d_in[6];
    float* out = (float*)d_out;

    (void)hipFuncSetAttribute(reinterpret_cast<const void*>(deform_conv1d_wmma),
                              hipFuncAttributeMaxDynamicSharedMemorySize, LDS_BYTES);

    dim3 grid(16 * CHUNKS);      // 256 blocks: 16 batches x 16 l-chunks
    dim3 block(THREADS);
    deform_conv1d_wmma<<<grid, block, LDS_BYTES, stream>>>(
        x, p_w, p_b, m_w, m_b, c_w, c_b, out);
}